// FasterRCNNSoftLabels_43198781063709
// MI455X (gfx1250) — compile-verified
//
#include <hip/hip_runtime.h>
#include <math.h>

#define BB 4
#define NN 2048
#define CC 3
#define MM (NN * (CC - 1))      // 4096 candidates per image
#define NDET 100
#define SCORE_THR 0.05f
#define NMS_THR 0.5f
#define MIN_SZ 0.01f
#define SCALE_CLAMP 4.135166556742356f   // log(1000/16)
#define NT 1024
#define CHUNK 32                 // one wave32 per NMS chunk

// ---------------------------------------------------------------- init
__global__ void frcnn_init_kernel(int* __restrict__ maxbits) {
    if (threadIdx.x < BB) maxbits[threadIdx.x] = 0;   // 0.0f bits; coords >= 0
}

// ---------------------------------------------------------------- decode
// One thread per (b, n): softmax over 3 logits, decode+clip the 2 foreground
// class boxes, validity + sort key, per-image coordinate max (atomic).
__global__ void frcnn_decode_kernel(const float* __restrict__ logits,
                                    const float* __restrict__ reg,
                                    const float* __restrict__ props,
                                    const int* __restrict__ ih,
                                    const int* __restrict__ iw,
                                    float* __restrict__ dec_boxes,   // [B][M][4]
                                    float* __restrict__ sortkey,     // [B][M]
                                    int* __restrict__ maxbits) {     // [B]
    int t = blockIdx.x * blockDim.x + threadIdx.x;
    if (t >= BB * NN) return;
    int b = t / NN, n = t % NN;

    const float Hc = (float)(*ih);
    const float Wc = (float)(*iw);

    const float* L = logits + (size_t)t * CC;
    const float* R = reg + (size_t)t * 4 * CC;
    __builtin_prefetch(R, 0, 0);   // global_prefetch_b8

    float l0 = L[0], l1 = L[1], l2 = L[2];
    float mxl = fmaxf(l0, fmaxf(l1, l2));
    float e0 = __expf(l0 - mxl), e1 = __expf(l1 - mxl), e2 = __expf(l2 - mxl);
    float inv = 1.0f / (e0 + e1 + e2);
    float scs[2] = { e1 * inv, e2 * inv };

    const float* P = props + (size_t)t * 4;
    float px1 = P[0], py1 = P[1], px2 = P[2], py2 = P[3];
    float w = px2 - px1, h = py2 - py1;
    float cx = px1 + 0.5f * w, cy = py1 + 0.5f * h;

    float lm = 0.0f;
#pragma unroll
    for (int c = 1; c < CC; ++c) {
        float dx = R[c * 4 + 0] * (1.0f / 10.0f);
        float dy = R[c * 4 + 1] * (1.0f / 10.0f);
        float dw = fminf(R[c * 4 + 2] * (1.0f / 5.0f), SCALE_CLAMP);
        float dh = fminf(R[c * 4 + 3] * (1.0f / 5.0f), SCALE_CLAMP);
        float pcx = dx * w + cx;
        float pcy = dy * h + cy;
        float pw = __expf(dw) * w;
        float ph = __expf(dh) * h;
        float x1 = fminf(fmaxf(pcx - 0.5f * pw, 0.0f), Wc);
        float y1 = fminf(fmaxf(pcy - 0.5f * ph, 0.0f), Hc);
        float x2 = fminf(fmaxf(pcx + 0.5f * pw, 0.0f), Wc);
        float y2 = fminf(fmaxf(pcy + 0.5f * ph, 0.0f), Hc);

        int m = n * (CC - 1) + (c - 1);
        float* D = dec_boxes + ((size_t)b * MM + m) * 4;
        D[0] = x1; D[1] = y1; D[2] = x2; D[3] = y2;

        float sc = scs[c - 1];
        bool valid = (sc > SCORE_THR) && ((x2 - x1) >= MIN_SZ) && ((y2 - y1) >= MIN_SZ);
        sortkey[(size_t)b * MM + m] = valid ? sc : -__builtin_inff();
        lm = fmaxf(lm, fmaxf(fmaxf(x1, y1), fmaxf(x2, y2)));
    }
    atomicMax(&maxbits[b], __float_as_int(lm));   // non-negative: int cmp == float cmp
}

// ---------------------------------------------------------------- sort
// Per-image bitonic sort of 4096 (key desc, payload = original index) in LDS.
__global__ __launch_bounds__(NT)
void frcnn_sort_kernel(const float* __restrict__ sortkey,
                       int* __restrict__ order) {
    __shared__ float kk[MM];
    __shared__ int   id[MM];
    int b = blockIdx.x;
    const float* SK = sortkey + (size_t)b * MM;

    for (int i = threadIdx.x; i < MM; i += NT) { kk[i] = SK[i]; id[i] = i; }
    __syncthreads();

    for (int sz = 2; sz <= MM; sz <<= 1) {
        for (int st = sz >> 1; st > 0; st >>= 1) {
            for (int i = threadIdx.x; i < MM; i += NT) {
                int p = i ^ st;
                if (p > i) {
                    bool desc = ((i & sz) == 0);
                    float ki = kk[i], kp = kk[p];
                    bool sw = desc ? (ki < kp) : (ki > kp);
                    if (sw) {
                        kk[i] = kp; kk[p] = ki;
                        int ti = id[i]; id[i] = id[p]; id[p] = ti;
                    }
                }
            }
            __syncthreads();
        }
    }
    int* O = order + (size_t)b * MM;
    for (int i = threadIdx.x; i < MM; i += NT) O[i] = id[i];
}

// ---------------------------------------------------------------- nms + emit
// Async-gather sorted boxes into LDS, add per-class offset, blocked greedy
// NMS (intra-chunk: one wave32 + ballot masks; cross-chunk: full block),
// then compact kept-first top-100 into d_out.
__global__ __launch_bounds__(NT)
void frcnn_nms_kernel(const float* __restrict__ dec_boxes,
                      const float* __restrict__ sortkey,
                      const int* __restrict__ order,
                      const int* __restrict__ maxbits,
                      float* __restrict__ out) {
    __shared__ float4        sbox[MM];    // 64 KB (of 320 KB WGP LDS pool)
    __shared__ unsigned char keepf[MM];   //  4 KB
    __shared__ unsigned int  chmask;      // kept-mask of the current chunk
    int b = blockIdx.x;
    const int*   O  = order + (size_t)b * MM;
    const float* SK = sortkey + (size_t)b * MM;
    float off_unit = __int_as_float(maxbits[b]) + 1.0f;

    // Gathered async copies: per-lane global VGPR address -> per-lane LDS
    // VGPR address, 16B each (ASYNCcnt-tracked, no VGPR round trip).
    for (int j = threadIdx.x; j < MM; j += NT) {
        int oi = O[j];
        unsigned long long ga =
            (unsigned long long)(const void*)(dec_boxes + ((size_t)b * MM + oi) * 4);
        unsigned int la = (unsigned int)(unsigned long long)(void*)&sbox[j];
        asm volatile("global_load_async_to_lds_b128 %0, %1, off"
                     :: "v"(la), "v"(ga) : "memory");
        keepf[j] = (SK[oi] > -__builtin_inff()) ? (unsigned char)1 : (unsigned char)0;
    }
    asm volatile("s_wait_asynccnt 0" ::: "memory");
    __syncthreads();

    // batched-NMS offset: boxes + label*(max+1), label = (origIdx & 1) + 1
    for (int j = threadIdx.x; j < MM; j += NT) {
        float off = (float)((O[j] & 1) + 1) * off_unit;
        float4 v = sbox[j];
        v.x += off; v.y += off; v.z += off; v.w += off;
        sbox[j] = v;
    }
    __syncthreads();

    // Blocked greedy NMS, rank order preserved (== reference fori_loop).
    for (int r0 = 0; r0 < MM; r0 += CHUNK) {
        // Phase 1: wave 0 resolves rows [r0, r0+32) sequentially; the live
        // mask lives in an SGPR via wave32 ballot (no LDS in the inner loop).
        if (threadIdx.x < CHUNK) {
            int l = threadIdx.x;
            float4 bl = sbox[r0 + l];
            float  al = (bl.z - bl.x) * (bl.w - bl.y);
            bool alive = (keepf[r0 + l] != 0);
            unsigned int km = __builtin_amdgcn_ballot_w32(alive);
            for (int i = 0; i < CHUNK - 1; ++i) {
                if (km & (1u << i)) {                 // uniform: row i still kept
                    float4 bi = sbox[r0 + i];         // LDS broadcast read
                    float  ai = (bi.z - bi.x) * (bi.w - bi.y);
                    float ix = fmaxf(fminf(bl.z, bi.z) - fmaxf(bl.x, bi.x), 0.0f);
                    float iy = fmaxf(fminf(bl.w, bi.w) - fmaxf(bl.y, bi.y), 0.0f);
                    float inter = ix * iy;
                    float iou = inter / (al + ai - inter + 1e-7f);
                    if (l > i && iou > NMS_THR) alive = false;
                    km = __builtin_amdgcn_ballot_w32(alive);
                }
            }
            keepf[r0 + l] = alive ? (unsigned char)1 : (unsigned char)0;
            if (l == 0) chmask = km;
        }
        __syncthreads();

        // Phase 2: every thread screens its j > r0+31 against the chunk's
        // kept rows (bit-scan over the mask).
        unsigned int km = chmask;
        if (km != 0u) {
            for (int j = r0 + CHUNK + (int)threadIdx.x; j < MM; j += NT) {
                if (keepf[j]) {
                    float4 bj = sbox[j];
                    float  aj = (bj.z - bj.x) * (bj.w - bj.y);
                    bool kill = false;
                    unsigned int m = km;
                    while (m) {
                        int i = __builtin_ctz(m);
                        m &= m - 1u;
                        float4 bi = sbox[r0 + i];
                        float  ai = (bi.z - bi.x) * (bi.w - bi.y);
                        float ix = fmaxf(fminf(bj.z, bi.z) - fmaxf(bj.x, bi.x), 0.0f);
                        float iy = fmaxf(fminf(bj.w, bi.w) - fmaxf(bj.y, bi.y), 0.0f);
                        float inter = ix * iy;
                        float iou = inter / (aj + ai - inter + 1e-7f);
                        kill |= (iou > NMS_THR);
                    }
                    if (kill) keepf[j] = 0;
                }
            }
        }
        __syncthreads();
    }

    // Emit: kept rows first (score order preserved), remainder zeroed.
    if (threadIdx.x == 0) {
        float* ob = out + (size_t)b * NDET * 4;
        float* os = out + (size_t)BB * NDET * 4 + (size_t)b * NDET;
        float* ol = out + (size_t)BB * NDET * 5 + (size_t)b * NDET;
        float* ok = out + (size_t)BB * NDET * 6 + (size_t)b * NDET;
        int cnt = 0;
        for (int r = 0; r < MM && cnt < NDET; ++r) {
            if (keepf[r]) {
                int oi = O[r];
                const float* g = dec_boxes + ((size_t)b * MM + oi) * 4;
                ob[cnt * 4 + 0] = g[0];
                ob[cnt * 4 + 1] = g[1];
                ob[cnt * 4 + 2] = g[2];
                ob[cnt * 4 + 3] = g[3];
                os[cnt] = SK[oi];
                ol[cnt] = (float)((oi & 1) + 1);
                ok[cnt] = 1.0f;
                ++cnt;
            }
        }
        for (; cnt < NDET; ++cnt) {
            ob[cnt * 4 + 0] = 0.0f; ob[cnt * 4 + 1] = 0.0f;
            ob[cnt * 4 + 2] = 0.0f; ob[cnt * 4 + 3] = 0.0f;
            os[cnt] = 0.0f; ol[cnt] = 0.0f; ok[cnt] = 0.0f;
        }
    }
}

// ---------------------------------------------------------------- launch
extern "C" void kernel_launch(void* const* d_in, const int* in_sizes, int n_in,
                              void* d_out, int out_size, void* d_ws, size_t ws_size,
                              hipStream_t stream) {
    (void)in_sizes; (void)n_in; (void)out_size; (void)ws_size;
    const float* logits = (const float*)d_in[0];   // [B,N,C]
    const float* reg    = (const float*)d_in[1];   // [B,N,4C]
    const float* props  = (const float*)d_in[2];   // [B,N,4]
    const int*   ih     = (const int*)d_in[3];
    const int*   iw     = (const int*)d_in[4];
    float* out = (float*)d_out;

    // workspace carve (~400 KB total)
    float* dec_boxes = (float*)d_ws;                          // B*M*4
    float* sortkey   = dec_boxes + (size_t)BB * MM * 4;       // B*M
    int*   order     = (int*)(sortkey + (size_t)BB * MM);     // B*M
    int*   maxbits   = order + (size_t)BB * MM;               // B

    frcnn_init_kernel<<<1, 64, 0, stream>>>(maxbits);
    frcnn_decode_kernel<<<(BB * NN + 255) / 256, 256, 0, stream>>>(
        logits, reg, props, ih, iw, dec_boxes, sortkey, maxbits);
    frcnn_sort_kernel<<<BB, NT, 0, stream>>>(sortkey, order);
    frcnn_nms_kernel<<<BB, NT, 0, stream>>>(dec_boxes, sortkey, order, maxbits, out);
}